// FIRST_BEAM_SEARCH_37391985279366
// MI455X (gfx1250) — compile-verified
//
#include <hip/hip_runtime.h>
#include <hip/hip_bf16.h>
#include <math.h>

// ---------------- problem geometry ----------------
#define LAYERS 32
#define HEADS  8
#define SEQ    1024
#define DIM    128
#define VOCAB  128000
#define BEAM   3
#define HIST   16

constexpr long long HSD        = (long long)HEADS * SEQ * DIM;       // 1,048,576 f32 per layer (4 MiB)
constexpr long long KV_ELEMS   = (long long)LAYERS * HSD;            // 33,554,432 f32 (128 MiB)
constexpr long long KVREP      = KV_ELEMS * BEAM;                    // 100,663,296
constexpr long long OFF_TOPIDX = KVREP;                              // +3
constexpr long long OFF_SAVEID = OFF_TOPIDX + BEAM;                  // +51
constexpr long long OFF_PEN    = OFF_SAVEID + (long long)BEAM * (HIST + 1);  // +384000
constexpr long long OFF_PROB   = OFF_PEN + (long long)BEAM * VOCAB;  // +3
constexpr long long OFF_MAXIDX = OFF_PROB + BEAM;                    // +1  => total 101,047,354

// ---------------- CDNA5 async global<->LDS path ----------------
#define AS1 __attribute__((address_space(1)))
#define AS3 __attribute__((address_space(3)))

// gfx12+ CPol immediate: TH in bits [2:0]; store TH: 0=RT, 1=NT (ISA cache-control §7.3)
#define CPOL_RT 0
#define CPOL_NT 1

typedef __attribute__((__vector_size__(4 * sizeof(int)))) int v4i;

#if defined(__has_builtin)
#  if __has_builtin(__builtin_amdgcn_global_load_async_to_lds_b128) && \
      __has_builtin(__builtin_amdgcn_global_store_async_from_lds_b128)
#    define HAVE_ASYNC_BUILTIN 1
#  endif
#  if __has_builtin(__builtin_amdgcn_s_wait_asynccnt)
#    define HAVE_WAIT_BUILTIN 1
#  endif
#endif

__device__ __forceinline__ void g2l_b128(const void* g, void* l) {
#if defined(__HIP_DEVICE_COMPILE__)
#  if defined(HAVE_ASYNC_BUILTIN)
  // TH=RT: let the kv source fill L2 so replayed reads hit the 192MB L2
  __builtin_amdgcn_global_load_async_to_lds_b128((AS1 v4i*)(void*)g, (AS3 v4i*)l, 0, CPOL_RT);
#  else
  unsigned la = (unsigned)(unsigned long long)(AS3 char*)(char*)l;
  asm volatile("global_load_async_to_lds_b128 %0, %1, off"
               :: "v"(la), "v"(g) : "memory");
#  endif
#else
  (void)g; (void)l;
#endif
}

__device__ __forceinline__ void l2g_b128_nt(void* g, const void* l) {
#if defined(__HIP_DEVICE_COMPILE__)
#  if defined(HAVE_ASYNC_BUILTIN)
  // TH=NT: 403MB write stream must not sweep the kv source out of L2
  __builtin_amdgcn_global_store_async_from_lds_b128((AS1 v4i*)g, (AS3 v4i*)(void*)l, 0, CPOL_NT);
#  else
  unsigned la = (unsigned)(unsigned long long)(AS3 char*)(char*)(void*)l;
  asm volatile("global_store_async_from_lds_b128 %0, %1, off th:TH_STORE_NT"
               :: "v"(g), "v"(la) : "memory");
#  endif
#else
  (void)g; (void)l;
#endif
}

__device__ __forceinline__ void wait_async0() {
#if defined(__HIP_DEVICE_COMPILE__)
#  if defined(HAVE_WAIT_BUILTIN)
  __builtin_amdgcn_s_wait_asynccnt(0);
#  else
  asm volatile("s_wait_asynccnt 0x0" ::: "memory");
#  endif
#endif
}

// ---------------- kernel 1: KV 3x replication (bandwidth bound) ----------------
#define TILE_BYTES 32768
#define CPY_TPB    256

__global__ __launch_bounds__(CPY_TPB)
void kv_replicate_kernel(const float* __restrict__ kv, float* __restrict__ out) {
  __shared__ alignas(16) unsigned char tile[TILE_BYTES];
  const long long layerBytes = HSD * 4;                        // 4 MiB, power of two
  const long long chunk  = (long long)blockIdx.x * TILE_BYTES; // byte offset into kv
  const long long layer  = chunk >> 22;
  const long long within = chunk & (layerBytes - 1);
  const char* src = (const char*)kv + chunk;
  const int t = threadIdx.x;

  // global -> LDS, 8x B128 per thread (thread t owns its own LDS bytes)
#pragma unroll
  for (int i = 0; i < TILE_BYTES / (CPY_TPB * 16); ++i) {
    const int off = t * 16 + i * (CPY_TPB * 16);
    g2l_b128(src + off, tile + off);
  }
  wait_async0();   // per-wave: our lanes' LDS bytes are resident

  // LDS -> global, fan out (non-temporal) to the 3 beam copies of this layer
#pragma unroll
  for (int b = 0; b < BEAM; ++b) {
    char* dst = (char*)out + (layer * BEAM + b) * layerBytes + within;
#pragma unroll
    for (int i = 0; i < TILE_BYTES / (CPY_TPB * 16); ++i) {
      const int off = t * 16 + i * (CPY_TPB * 16);
      l2g_b128_nt(dst + off, tile + off);
    }
  }
  wait_async0();
}

// ---------------- kernel 2: log_softmax + top-3 + small outputs ----------------
#define TPB2 1024

__device__ __forceinline__ bool beats(float xv, int xi, float yv, int yi) {
  return (xv > yv) || (xv == yv && xi < yi);   // jax top_k tie-break: lower index first
}

__global__ __launch_bounds__(TPB2)
void topk_kernel(const float* __restrict__ logits, const int* __restrict__ save_id,
                 float* __restrict__ out, int* __restrict__ ws_idx) {
  __shared__ float sm[TPB2], ss[TPB2];
  __shared__ float stv[TPB2 * BEAM];
  __shared__ int   sti[TPB2 * BEAM];

  const int t = threadIdx.x;
  float m = -INFINITY, s = 0.f;
  float tv0 = -INFINITY, tv1 = -INFINITY, tv2 = -INFINITY;
  int   ti0 = 0x7fffffff, ti1 = 0x7fffffff, ti2 = 0x7fffffff;

  for (int v = t; v < VOCAB; v += TPB2) {
    const float x = logits[v];
    if (x > m) { s = s * expf(m - x) + 1.f; m = x; }   // online softmax accumulate
    else       { s += expf(x - m); }
    if (beats(x, v, tv2, ti2)) {
      if (beats(x, v, tv1, ti1)) {
        tv2 = tv1; ti2 = ti1;
        if (beats(x, v, tv0, ti0)) { tv1 = tv0; ti1 = ti0; tv0 = x; ti0 = v; }
        else                       { tv1 = x;  ti1 = v; }
      } else { tv2 = x; ti2 = v; }
    }
  }
  sm[t] = m; ss[t] = s;
  stv[t * 3 + 0] = tv0; sti[t * 3 + 0] = ti0;
  stv[t * 3 + 1] = tv1; sti[t * 3 + 1] = ti1;
  stv[t * 3 + 2] = tv2; sti[t * 3 + 2] = ti2;
  __syncthreads();

  if (t == 0) {
    float M = -INFINITY;
    for (int i = 0; i < TPB2; ++i) M = fmaxf(M, sm[i]);
    float S = 0.f;
    for (int i = 0; i < TPB2; ++i) S += ss[i] * expf(sm[i] - M);

    float gv0 = -INFINITY, gv1 = -INFINITY, gv2 = -INFINITY;
    int   gi0 = 0x7fffffff, gi1 = 0x7fffffff, gi2 = 0x7fffffff;
    for (int i = 0; i < TPB2 * BEAM; ++i) {
      const float x = stv[i]; const int v = sti[i];
      if (beats(x, v, gv2, gi2)) {
        if (beats(x, v, gv1, gi1)) {
          gv2 = gv1; gi2 = gi1;
          if (beats(x, v, gv0, gi0)) { gv1 = gv0; gi1 = gi0; gv0 = x; gi0 = v; }
          else                       { gv1 = x;  gi1 = v; }
        } else { gv2 = x; gi2 = v; }
      }
    }
    const float logZ = M + logf(S);
    const int   gi[BEAM] = {gi0, gi1, gi2};
    const float gv[BEAM] = {gv0, gv1, gv2};
    for (int b = 0; b < BEAM; ++b) {
      out[OFF_TOPIDX + b] = (float)gi[b];
      out[OFF_PROB + b]   = gv[b] - logZ;
      ws_idx[b] = gi[b];
    }
    out[OFF_MAXIDX] = (float)gi0;
    for (int r = 0; r < BEAM; ++r) {
      for (int c = 0; c < HIST; ++c)
        out[OFF_SAVEID + r * (HIST + 1) + c] = (float)save_id[r * HIST + c];
      out[OFF_SAVEID + r * (HIST + 1) + HIST] = (float)gi[r];
    }
  }
}

// ---------------- kernel 3: repetition-penalty copy + 3-element scatter ----------------
__global__ __launch_bounds__(256)
void penalty_kernel(const float* __restrict__ pen, const float* __restrict__ pv,
                    const int* __restrict__ ws_idx, float* __restrict__ out) {
  const int e = blockIdx.x * 256 + threadIdx.x;
  if (e >= BEAM * VOCAB) return;
  const int row = e / VOCAB;
  const int col = e - row * VOCAB;
  float x = pen[e];
  if (col == ws_idx[row]) x *= pv[0];
  __builtin_nontemporal_store(x, &out[OFF_PEN + e]);
}

// ---------------- launcher ----------------
extern "C" void kernel_launch(void* const* d_in, const int* in_sizes, int n_in,
                              void* d_out, int out_size, void* d_ws, size_t ws_size,
                              hipStream_t stream) {
  const float* kv      = (const float*)d_in[0];
  const float* logits  = (const float*)d_in[1];
  const int*   save_id = (const int*)d_in[2];
  const float* pen     = (const float*)d_in[3];
  const float* pv      = (const float*)d_in[4];
  float* out = (float*)d_out;
  int*   ws_idx = (int*)d_ws;
  (void)in_sizes; (void)n_in; (void)out_size; (void)ws_size;

  topk_kernel<<<1, TPB2, 0, stream>>>(logits, save_id, out, ws_idx);
  penalty_kernel<<<(BEAM * VOCAB + 255) / 256, 256, 0, stream>>>(pen, pv, ws_idx, out);
  const int nblocks = (int)((KV_ELEMS * 4) / TILE_BYTES);   // 4096 blocks x 32 KiB
  kv_replicate_kernel<<<nblocks, CPY_TPB, 0, stream>>>(kv, out);
}